// GraphPoolOut_90434831384776
// MI455X (gfx1250) — compile-verified
//
#include <hip/hip_runtime.h>

// Voxel scatter-max pooling for MI455X (gfx1250, wave32).
//
// Strategy (memory-bound, no matrix math possible):
//  1) vox_init:        write sentinel 0 over the 64 MB output (uint4 stores).
//  2) vox_scatter_max: one wave32 per point. Lanes load the 64-ch feature row
//     (non-temporal: streamed once, keep L2 for the atomic working set),
//     encode floats into an order-preserving u32, and issue
//     global_atomic_max_u32 into the voxel row (2 x 128B contiguous bursts).
//  3) vox_finalize:    decode in place; sentinel 0 -> 0.0f (empty voxel).
//
// Order-preserving encode: for f >= 0: bits | 0x80000000 ; for f < 0: ~bits.
// Monotone in float order, every finite float encodes > 0, so u32 max == f32 max
// and 0 is a safe "empty" identity. Output (64 MB) stays resident in the
// 192 MB L2, so the 64M atomics never leave the chip.
//
// NOTE: vox_scatter_max is placed first in the file so its disassembly shows
// in the compile-feedback snippet (verifying global_atomic_max_u32 + NT hints).

namespace {

__device__ __forceinline__ unsigned encf(float f) {
  unsigned u = __float_as_uint(f);
  return (u & 0x80000000u) ? ~u : (u | 0x80000000u);
}

__device__ __forceinline__ float decf(unsigned u) {
  if (u == 0u) return 0.0f;  // empty-voxel sentinel
  return (u & 0x80000000u) ? __uint_as_float(u & 0x7FFFFFFFu)
                           : __uint_as_float(~u);
}

}  // namespace

__global__ void vox_scatter_max(const int* __restrict__ verts,
                                const float* __restrict__ feats,
                                unsigned* __restrict__ out, int npts) {
  const int lane = threadIdx.x & 31;
  const int wid  = blockIdx.x * (blockDim.x >> 5) + (threadIdx.x >> 5);
  if (wid >= npts) return;

  // Wave-uniform point coords; streamed once -> non-temporal (one cacheline
  // request per wave after coalescing).
  const int* vp = verts + (size_t)3 * (size_t)wid;
  const int x = __builtin_nontemporal_load(vp + 0) >> 2;  // POOL_SIZE = 4
  const int y = __builtin_nontemporal_load(vp + 1) >> 2;
  const int z = __builtin_nontemporal_load(vp + 2) >> 2;
  const unsigned vox = (unsigned)((x << 12) + (y << 6) + z);  // GRID = 64

  // 64-channel feature row: lanes cover [lane] and [lane+32].
  const float* fp = feats + ((size_t)wid << 6);
  const float a = __builtin_nontemporal_load(fp + lane);
  const float b = __builtin_nontemporal_load(fp + lane + 32);

  // Each atomic instruction from this wave covers one contiguous 128 B span
  // of the 256 B voxel row -> fully coalesced L2 atomics. Results unused ->
  // no-return atomics (STOREcnt), half the atomic return traffic.
  unsigned* o = out + ((size_t)vox << 6);
  atomicMax(o + lane,      encf(a));
  atomicMax(o + lane + 32, encf(b));
}

__global__ void vox_init(uint4* __restrict__ out, int n4) {
  int i = blockIdx.x * blockDim.x + threadIdx.x;
  if (i < n4) out[i] = uint4{0u, 0u, 0u, 0u};
}

__global__ void vox_finalize(unsigned* __restrict__ buf, int n4) {
  int i = blockIdx.x * blockDim.x + threadIdx.x;
  if (i >= n4) return;
  uint4 u = ((const uint4*)buf)[i];
  float4 f{decf(u.x), decf(u.y), decf(u.z), decf(u.w)};
  ((float4*)buf)[i] = f;
}

extern "C" void kernel_launch(void* const* d_in, const int* in_sizes, int n_in,
                              void* d_out, int out_size, void* d_ws, size_t ws_size,
                              hipStream_t stream) {
  (void)n_in; (void)d_ws; (void)ws_size;

  const int*   verts = (const int*)d_in[0];   // [N,3] voxel coords in [0,256)
  const float* feats = (const float*)d_in[1]; // [N,64] f32 features
  unsigned*    out   = (unsigned*)d_out;      // [262144*64] reinterpreted u32

  const int npts = in_sizes[0] / 3;  // 1,000,000
  const int n4   = out_size / 4;     // 4,194,304 uint4's

  // 1) sentinel init (64 MB of zeros, vectorized)
  vox_init<<<(n4 + 255) / 256, 256, 0, stream>>>((uint4*)out, n4);

  // 2) scatter-max: one wave32 per point, 8 waves per 256-thread block
  const int wavesPerBlock = 256 / 32;
  vox_scatter_max<<<(npts + wavesPerBlock - 1) / wavesPerBlock, 256, 0, stream>>>(
      verts, feats, out, npts);

  // 3) decode in place; empty voxels -> 0.0f
  vox_finalize<<<(n4 + 255) / 256, 256, 0, stream>>>(out, n4);
}